// SimpleRnnNetwork_25503515804146
// MI455X (gfx1250) — compile-verified
//
#include <hip/hip_runtime.h>

typedef __attribute__((ext_vector_type(2))) float v2f;
typedef __attribute__((ext_vector_type(8))) float v8f;

// CDNA5 hardware tanh (TRANS32). Fall back to inline asm if the builtin
// name differs in this toolchain.
__device__ __forceinline__ float fast_tanh(float x) {
#if __has_builtin(__builtin_amdgcn_tanhf)
    return __builtin_amdgcn_tanhf(x);
#elif __has_builtin(__builtin_amdgcn_tanh_f32)
    return __builtin_amdgcn_tanh_f32(x);
#else
    float y;
    asm("v_tanh_f32 %0, %1" : "=v"(y) : "v"(x));
    return y;
#endif
}

__device__ __forceinline__ float fast_sigmoid(float x) {
    // sigmoid(x) = 0.5*tanh(0.5x) + 0.5  -> single transcendental
    return fmaf(0.5f, fast_tanh(0.5f * x), 0.5f);
}

// ---------------------------------------------------------------------------
// Phase 1: parallel precompute of per-step gate pre-activations
//   a[t][n] = w_ih[n]*x[t] + b_ih[n] + b_hh[n],  n in {i,f,g,o}
// Rank-1 GEMM tile per wave via V_WMMA_F32_16X16X4_F32:
//   A (16x4): column K=0 holds x[t0..t0+15], rest zero
//   B (4x16): row    K=0 holds w_ih[n] for n<4, rest zero
//   C (16x16): column n holds bias[n] (broadcast over rows)
// ---------------------------------------------------------------------------
__global__ void lstm_pre_wmma(const float* __restrict__ x,
                              const float* __restrict__ w_ih,
                              const float* __restrict__ b_ih,
                              const float* __restrict__ b_hh,
                              float* __restrict__ a_out, int T) {
    const int lane = threadIdx.x & 31;
    const int wave = (int)((blockIdx.x * blockDim.x + threadIdx.x) >> 5);
    const int t0   = wave * 16;
    if (t0 >= T) return;  // uniform per wave -> EXEC all-1 inside

    const int n = lane & 15;

    // A operand: only K=0 column nonzero (x values, lanes 0-15 of VGPR0).
    int xi = t0 + n; if (xi >= T) xi = T - 1;
    float xv = x[xi];
    v2f a; a.x = (lane < 16) ? xv : 0.0f; a.y = 0.0f;

    // B operand: only K=0 row nonzero: w_ih[n] for n<4.
    float wv = w_ih[n & 3];
    v2f b; b.x = (lane < 16 && n < 4) ? wv : 0.0f; b.y = 0.0f;

    // C operand: bias broadcast down each used column.
    float bias = b_ih[n & 3] + b_hh[n & 3];
    float cv = (n < 4) ? bias : 0.0f;
    v8f cm;
#pragma unroll
    for (int j = 0; j < 8; ++j) cm[j] = cv;

    // (neg_a, A, neg_b, B, c_mod, C, reuse_a, reuse_b)
    v8f d = __builtin_amdgcn_wmma_f32_16x16x4_f32(
        false, a, false, b, (short)0, cm, false, false);

    // D layout: VGPR j -> M = j (lanes 0-15) or j+8 (lanes 16-31); N = lane&15.
    const int half = lane >> 4;
    const int tb   = t0 + half * 8;
    if (t0 + 16 <= T) {
        // Full tile (always taken for T % 16 == 0): straight-line stores.
        if (n < 4) {
#pragma unroll
            for (int j = 0; j < 8; ++j) {
                a_out[(tb + j) * 4 + n] = d[j];
            }
        }
    } else {
        // Ragged last tile: per-element bounds check.
        if (n < 4) {
#pragma unroll
            for (int j = 0; j < 8; ++j) {
                int t = tb + j;
                if (t < T) a_out[t * 4 + n] = d[j];
            }
        }
    }
}

// ---------------------------------------------------------------------------
// Phase 2: the inherently-serial scan. One wave, all lanes redundantly compute
// the same chain (no divergence); lane 0 writes the result.
// Per step: 4 FMA (w_hh*h) + 5 v_tanh_f32 + 4 FMA + 5 MUL.
// Depth-8 register ring buffer prefetches pre-activations ~8 steps ahead so
// the (L2-resident) loads never sit on the dependent chain.
// ---------------------------------------------------------------------------
__global__ void lstm_scan(const float4* __restrict__ a4,
                          const float* __restrict__ w_hh,
                          const float* __restrict__ w_lin,
                          const float* __restrict__ b_lin,
                          float* __restrict__ out, int T) {
    const float w0 = w_hh[0], w1 = w_hh[1], w2 = w_hh[2], w3 = w_hh[3];
    float h = 0.0f, c = 0.0f;

    constexpr int PF = 8;
    float4 buf[PF];
#pragma unroll
    for (int k = 0; k < PF; ++k) buf[k] = a4[k < T ? k : 0];

    const int Tm = T & ~(PF - 1);
    for (int t = 0; t < Tm; t += PF) {
#pragma unroll
        for (int k = 0; k < PF; ++k) {
            float4 cur = buf[k];
            int nt = t + PF + k;
            if (nt < T) buf[k] = a4[nt];   // refill slot: consumed PF steps later

            float ig = fast_sigmoid(fmaf(w0, h, cur.x));
            float fg = fast_sigmoid(fmaf(w1, h, cur.y));
            float gg = fast_tanh  (fmaf(w2, h, cur.z));
            float og = fast_sigmoid(fmaf(w3, h, cur.w));
            c = fmaf(fg, c, ig * gg);
            h = og * fast_tanh(c);
        }
    }
    for (int t = Tm; t < T; ++t) {  // tail (unused for T=131072)
        float4 cur = a4[t];
        float ig = fast_sigmoid(fmaf(w0, h, cur.x));
        float fg = fast_sigmoid(fmaf(w1, h, cur.y));
        float gg = fast_tanh  (fmaf(w2, h, cur.z));
        float og = fast_sigmoid(fmaf(w3, h, cur.w));
        c = fmaf(fg, c, ig * gg);
        h = og * fast_tanh(c);
    }

    if (threadIdx.x == 0) out[0] = fmaf(w_lin[0], h, b_lin[0]);
}

extern "C" void kernel_launch(void* const* d_in, const int* in_sizes, int n_in,
                              void* d_out, int out_size, void* d_ws, size_t ws_size,
                              hipStream_t stream) {
    const float* x     = (const float*)d_in[0];
    const float* w_ih  = (const float*)d_in[1];
    const float* w_hh  = (const float*)d_in[2];
    const float* b_ih  = (const float*)d_in[3];
    const float* b_hh  = (const float*)d_in[4];
    const float* w_lin = (const float*)d_in[5];
    const float* b_lin = (const float*)d_in[6];
    float* out = (float*)d_out;
    const int T = in_sizes[0];

    float* a = (float*)d_ws;  // T*4 floats = 2 MB of pre-activations

    const int tiles = (T + 15) / 16;          // one wave per 16-timestep tile
    const int wavesPerBlock = 8;              // 256 threads = 8 wave32
    const int blocks = (tiles + wavesPerBlock - 1) / wavesPerBlock;

    lstm_pre_wmma<<<blocks, wavesPerBlock * 32, 0, stream>>>(
        x, w_ih, b_ih, b_hh, a, T);
    lstm_scan<<<1, 32, 0, stream>>>(
        (const float4*)a, w_hh, w_lin, b_lin, out, T);
}